// MultiEdgeAggModule_36249523978495
// MI455X (gfx1250) — compile-verified
//
#include <hip/hip_runtime.h>
#include <hip/hip_bf16.h>

typedef __attribute__((ext_vector_type(2))) float v2f;
typedef __attribute__((ext_vector_type(4))) float v4f;
typedef __attribute__((ext_vector_type(8))) float v8f;

#define HDIM 128
#define BM 80            // rows per block  (250000 % 80 == 0 -> no tail)
#define NT 5             // 16-row M tiles per block
#define STRIDE 132       // LDS row stride (floats): 528B = 16B-aligned rows,
                         // word stride 132 % 64 == 4 -> conflict-free ds_load_b64

// ---------------------------------------------------------------- zero scratch
__global__ void zero_f32(float* __restrict__ p, long n) {
  long i = (long)blockIdx.x * blockDim.x + threadIdx.x;
  if (i < n) p[i] = 0.0f;
}

// ------------------------------------------------- segment-sum scatter (atomics)
// 32 threads/edge; non-temporal float4 streaming of edge_attr (read-once 512MB),
// atomic f32 accumulation into agg (128MB, L2-resident).
// inv == simp_edge_batch (unique of values covering all of 0..S-1 is arange(S)).
__global__ void scatter_kernel(const int* __restrict__ edge_index,
                               const float* __restrict__ edge_attr,
                               const int* __restrict__ simp,
                               float* __restrict__ agg,
                               float* __restrict__ ei_sum,  // [2*S]
                               float* __restrict__ cnt,     // [S]
                               float* __restrict__ out_inv, // [E]
                               int E, int S) {
  long idx = (long)blockIdx.x * blockDim.x + threadIdx.x;
  int e  = (int)(idx >> 5);
  int c4 = (int)(idx & 31);
  if (e >= E) return;
  int g = simp[e];
  v4f v = __builtin_nontemporal_load(((const v4f*)(edge_attr + (long)e * HDIM)) + c4);
  float* dst = agg + (long)g * HDIM + c4 * 4;
  atomicAdd(dst + 0, v.x);
  atomicAdd(dst + 1, v.y);
  atomicAdd(dst + 2, v.z);
  atomicAdd(dst + 3, v.w);
  if (c4 == 0) {
    atomicAdd(&ei_sum[g],     (float)edge_index[e]);       // edge_index[0][e]
    atomicAdd(&ei_sum[S + g], (float)edge_index[E + e]);   // edge_index[1][e]
    atomicAdd(&cnt[g], 1.0f);
    out_inv[e] = (float)g;                                  // inv output
  }
}

// ------------------------------------------------------------- edge-index mean
__global__ void finalize_ei(const float* __restrict__ ei_sum,
                            const float* __restrict__ cnt,
                            float* __restrict__ out_ei, int S) {
  int g = blockIdx.x * blockDim.x + threadIdx.x;
  if (g >= S) return;
  float c = fmaxf(cnt[g], 1.0f);
  out_ei[g]     = ei_sum[g] / c;
  out_ei[S + g] = ei_sum[S + g] / c;
}

// --------------------------------------------- fused MLP: relu(agg@W1+b1)@W2+b2
// One block = 80 rows (5 M-tiles); wave w owns N-tile [16w, 16w+16).
// agg panel staged once into LDS (kills 8x inter-wave A redundancy); the SAME
// LDS buffer is reused for the h tile (GEMM1 consumes it fully first).
// V_WMMA_F32_16X16X4_F32 layouts (ISA 7.12.2):
//   A (16x4): lanes 0-15 -> M=lane, v2f={K0,K1}; lanes 16-31 -> M=lane-16, {K2,K3}
//   B (4x16): lanes 0-15 -> N=lane, v2f={K0,K1}; lanes 16-31 -> N=lane-16, {K2,K3}
//   C/D (16x16): element j of v8f at row j + 8*(lane>=16), col lane&15
__global__ __launch_bounds__(256)
void mlp_wmma(const float* __restrict__ agg,
              const float* __restrict__ W1, const float* __restrict__ b1,
              const float* __restrict__ W2, const float* __restrict__ b2,
              float* __restrict__ out, int S) {
  __shared__ float tile[BM * STRIDE];   // 42.2 KB, reused for A panel then h

  const int wave  = threadIdx.x >> 5;
  const int lane  = threadIdx.x & 31;
  const int m0    = blockIdx.x * BM;
  const int n0    = wave * 16;
  const int t     = lane & 15;          // A row / B,C,D column within tile
  const int kh    = (lane >> 4) * 2;    // K sub-offset: 0 (lanes 0-15) / 2 (16-31)
  const int rbase = (lane >> 4) * 8;    // C/D row base

  // ---- cooperative stage: agg[m0 : m0+BM, :] -> LDS (b128 loads/stores)
  for (int i = threadIdx.x; i < BM * (HDIM / 4); i += 256) {
    int r = i >> 5, c = i & 31;
    int row = m0 + r; if (row > S - 1) row = S - 1;   // clamp (no-op when 80|S)
    v4f v = *(((const v4f*)(agg + (long)row * HDIM)) + c);
    *((v4f*)&tile[r * STRIDE + c * 4]) = v;
  }
  __syncthreads();

  // ---- GEMM1: h = relu(A @ W1 + b1), A from LDS, 5 WMMAs per B load
  v8f acc[NT];
#pragma unroll
  for (int mt = 0; mt < NT; ++mt) acc[mt] = (v8f){};
#pragma unroll 4
  for (int k0 = 0; k0 < HDIM; k0 += 4) {
    v2f b; b.x = W1[(k0 + kh) * HDIM + n0 + t];
           b.y = W1[(k0 + kh + 1) * HDIM + n0 + t];
#pragma unroll
    for (int mt = 0; mt < NT; ++mt) {
      v2f a = *((const v2f*)&tile[(mt * 16 + t) * STRIDE + k0 + kh]);
      acc[mt] = __builtin_amdgcn_wmma_f32_16x16x4_f32(false, a, false, b,
                                                      (short)0, acc[mt],
                                                      false, false);
    }
  }
  __syncthreads();   // all A-panel reads complete before overwriting tile with h

  float bias1 = b1[n0 + t];
#pragma unroll
  for (int mt = 0; mt < NT; ++mt) {
#pragma unroll
    for (int j = 0; j < 8; ++j) {
      float v = acc[mt][j] + bias1;
      tile[(mt * 16 + rbase + j) * STRIDE + n0 + t] = v > 0.0f ? v : 0.0f;
    }
  }
  __syncthreads();

  // ---- GEMM2: out = h @ W2 + b2, A (= h) from LDS
  v8f acc2[NT];
#pragma unroll
  for (int mt = 0; mt < NT; ++mt) acc2[mt] = (v8f){};
#pragma unroll 4
  for (int k0 = 0; k0 < HDIM; k0 += 4) {
    v2f b; b.x = W2[(k0 + kh) * HDIM + n0 + t];
           b.y = W2[(k0 + kh + 1) * HDIM + n0 + t];
#pragma unroll
    for (int mt = 0; mt < NT; ++mt) {
      v2f a = *((const v2f*)&tile[(mt * 16 + t) * STRIDE + k0 + kh]);
      acc2[mt] = __builtin_amdgcn_wmma_f32_16x16x4_f32(false, a, false, b,
                                                       (short)0, acc2[mt],
                                                       false, false);
    }
  }

  float bias2 = b2[n0 + t];
#pragma unroll
  for (int mt = 0; mt < NT; ++mt) {
#pragma unroll
    for (int j = 0; j < 8; ++j) {
      int row = m0 + mt * 16 + rbase + j;
      if (row < S) out[(long)row * HDIM + n0 + t] = acc2[mt][j] + bias2;
    }
  }
}

// ------------------------------------------------------------------- launcher
extern "C" void kernel_launch(void* const* d_in, const int* in_sizes, int n_in,
                              void* d_out, int out_size, void* d_ws, size_t ws_size,
                              hipStream_t stream) {
  const int*   edge_index = (const int*)d_in[0];
  const float* edge_attr  = (const float*)d_in[1];
  const int*   simp       = (const int*)d_in[2];
  const float* W1         = (const float*)d_in[3];
  const float* b1         = (const float*)d_in[4];
  const float* W2         = (const float*)d_in[5];
  const float* b2         = (const float*)d_in[6];

  const int E = in_sizes[2];
  // out_size = 2*S (new_edge_index) + S*H (new_edge_attr) + E (inv)
  const int S = (int)(((long)out_size - E) / (2 + HDIM));

  float* agg    = (float*)d_ws;           // S*H
  float* ei_sum = agg + (long)S * HDIM;   // 2*S
  float* cnt    = ei_sum + 2L * S;        // S

  float* out_ei   = (float*)d_out;
  float* out_attr = out_ei + 2L * S;
  float* out_inv  = out_attr + (long)S * HDIM;

  long nz = (long)S * HDIM + 3L * S;
  zero_f32<<<(int)((nz + 255) / 256), 256, 0, stream>>>(agg, nz);

  long nthr = (long)E * 32;
  scatter_kernel<<<(int)((nthr + 255) / 256), 256, 0, stream>>>(
      edge_index, edge_attr, simp, agg, ei_sum, cnt, out_inv, E, S);

  finalize_ei<<<(S + 255) / 256, 256, 0, stream>>>(ei_sum, cnt, out_ei, S);

  int mblocks = (S + BM - 1) / BM;        // 3125 for S = 250000
  mlp_wmma<<<mblocks, 256, 0, stream>>>(agg, W1, b1, W2, b2, out_attr, S);
}